// SimpleGraphAttentionLayer_34110630264945
// MI455X (gfx1250) — compile-verified
//
#include <hip/hip_runtime.h>
#include <hip/hip_bf16.h>

typedef float v2f __attribute__((ext_vector_type(2)));
typedef float v4f __attribute__((ext_vector_type(4)));
typedef float v8f __attribute__((ext_vector_type(8)));

#define BS 8
#define N 512
#define D 128
#define E 16
#define ASTRIDE 516   // 512 + 4 pad -> conflict-free LDS column access
#define HPSTRIDE 132  // 128 + 4 pad

// -------------------------------------------------------------------------
// Kernel 1: h = x @ W^T  (f32 WMMA 16x16x4).  One wave per 16-row tile.
// Also emits transposed copy hT[d][j] for fast B-frag loads in k_attn.
// A layout (16x4 f32): lanes 0-15 -> M=lane, K=0/1 in v0/v1;
//                      lanes 16-31 -> M=lane-16, K=2/3.
// B layout (4x16 f32): lanes 0-15 -> N=lane, K=0/1; lanes 16-31 -> K=2/3.
// Loads are hoisted into a separate unrolled loop so the 8 B-frags issue
// back-to-back (distinct regs) instead of serializing on one wait each.
// -------------------------------------------------------------------------
__global__ void k_h_gemm(const float* __restrict__ x,
                         const float* __restrict__ W,
                         float* __restrict__ h,
                         float* __restrict__ hT) {
  const int b    = blockIdx.x >> 5;
  const int m0   = (blockIdx.x & 31) << 4;
  const int lane = threadIdx.x & 31;
  const int half = lane >> 4;
  const int l16  = lane & 15;

  const float* xrow  = x + (size_t)(b * N + m0 + l16) * D + 2 * half;
  const float* wbase = W + (size_t)l16 * D + 2 * half;

  v8f c[8] = {};
  for (int kk = 0; kk < D; kk += 4) {
    v2f a = *(const v2f*)(xrow + kk);
    v2f bf[8];
#pragma unroll
    for (int f = 0; f < 8; ++f)
      bf[f] = *(const v2f*)(wbase + (size_t)(f * 16) * D + kk);
#pragma unroll
    for (int f = 0; f < 8; ++f)
      c[f] = __builtin_amdgcn_wmma_f32_16x16x4_f32(
          false, a, false, bf[f], (short)0, c[f], false, false);
  }

  // C layout: vgpr j, lanes 0-15 -> row m0+j, col 16f+lane; lanes 16-31 -> row m0+8+j
#pragma unroll
  for (int f = 0; f < 8; ++f) {
    float* hb = h  + (size_t)(b * N + m0 + 8 * half) * D + f * 16 + l16;
    float* tb = hT + (size_t)b * D * N + (size_t)(f * 16 + l16) * N + m0 + 8 * half;
#pragma unroll
    for (int j = 0; j < 8; ++j) {
      hb[(size_t)j * D] = c[f][j];
      tb[j]             = c[f][j];
    }
  }
}

// -------------------------------------------------------------------------
// Kernel 2: s_i = h . a_src ; s_j = h . a_dst  (one wave per row)
// -------------------------------------------------------------------------
__global__ void k_scores(const float* __restrict__ h,
                         const float* __restrict__ a_src,
                         const float* __restrict__ a_dst,
                         float* __restrict__ s_i,
                         float* __restrict__ s_j) {
  const int row  = blockIdx.x * 8 + (threadIdx.x >> 5);
  const int lane = threadIdx.x & 31;

  v4f hv = ((const v4f*)(h + (size_t)row * D))[lane];
  v4f av = ((const v4f*)a_src)[lane];
  v4f dv = ((const v4f*)a_dst)[lane];

  float si = hv.x * av.x + hv.y * av.y + hv.z * av.z + hv.w * av.w;
  float sj = hv.x * dv.x + hv.y * dv.y + hv.z * dv.z + hv.w * dv.w;
#pragma unroll
  for (int off = 16; off >= 1; off >>= 1) {
    si += __shfl_xor(si, off, 32);
    sj += __shfl_xor(sj, off, 32);
  }
  if (lane == 0) { s_i[row] = si; s_j[row] = sj; }
}

// -------------------------------------------------------------------------
// Kernel 3 (fused, HBM-bound): per (b, 16-row i-block):
//   e_ij = (s_i + s_j + edge.a_e)*scale  (edge streamed non-temporal, once)
//   row softmax -> alpha in LDS
//   h' = alpha @ h  via f32 WMMA, K unrolled x4, B-frags from transposed hT
//   LayerNorm(h') -> out
// -------------------------------------------------------------------------
__global__ void __launch_bounds__(256)
k_attn(const float* __restrict__ edge,
       const float* __restrict__ hT,
       const float* __restrict__ s_i,
       const float* __restrict__ s_j,
       const float* __restrict__ a_e,
       const float* __restrict__ gamma,
       const float* __restrict__ beta,
       float* __restrict__ out) {
  __shared__ float alpha[16 * ASTRIDE];   // 33 KB
  __shared__ float hp[16 * HPSTRIDE];     //  8.25 KB

  const int b   = blockIdx.y;
  const int m0  = blockIdx.x << 4;
  const int tid = threadIdx.x;
  const int r   = tid >> 4;   // 0..15 : row within i-block
  const int s   = tid & 15;   // 0..15 : column slot

  // ---- Phase 1: scores + softmax into LDS ----
  const v4f ae0 = ((const v4f*)a_e)[0];
  const v4f ae1 = ((const v4f*)a_e)[1];
  const v4f ae2 = ((const v4f*)a_e)[2];
  const v4f ae3 = ((const v4f*)a_e)[3];
  const float  si    = s_i[b * N + m0 + r];
  const float* sjrow = s_j + b * N;
  const float  scale = 0.088388347648318447f;  // 1/sqrt(128)
  const float* erow  = edge + ((size_t)(b * N + m0 + r) * N) * E;

  float ev[32];
  float mx = -3.4e38f;
#pragma unroll
  for (int t = 0; t < 32; ++t) {
    const int j = s + (t << 4);
    const v4f* ep = (const v4f*)(erow + (size_t)j * E);
    v4f e0 = __builtin_nontemporal_load(ep);
    v4f e1 = __builtin_nontemporal_load(ep + 1);
    v4f e2 = __builtin_nontemporal_load(ep + 2);
    v4f e3 = __builtin_nontemporal_load(ep + 3);
    float d;
    d  = e0.x * ae0.x + e0.y * ae0.y + e0.z * ae0.z + e0.w * ae0.w;
    d += e1.x * ae1.x + e1.y * ae1.y + e1.z * ae1.z + e1.w * ae1.w;
    d += e2.x * ae2.x + e2.y * ae2.y + e2.z * ae2.z + e2.w * ae2.w;
    d += e3.x * ae3.x + e3.y * ae3.y + e3.z * ae3.z + e3.w * ae3.w;
    float e = (si + sjrow[j] + d) * scale;
    ev[t] = e;
    mx = fmaxf(mx, e);
  }
#pragma unroll
  for (int off = 8; off >= 1; off >>= 1) mx = fmaxf(mx, __shfl_xor(mx, off, 16));
  float sum = 0.f;
#pragma unroll
  for (int t = 0; t < 32; ++t) { float a = __expf(ev[t] - mx); ev[t] = a; sum += a; }
#pragma unroll
  for (int off = 8; off >= 1; off >>= 1) sum += __shfl_xor(sum, off, 16);
  const float inv = 1.0f / sum;
#pragma unroll
  for (int t = 0; t < 32; ++t) alpha[r * ASTRIDE + s + (t << 4)] = ev[t] * inv;
  __syncthreads();

  // ---- Phase 2: h' tile = alpha @ h  (wave w -> cols [16w, 16w+16)) ----
  const int w    = tid >> 5;
  const int lane = tid & 31;
  const int half = lane >> 4;
  const int l16  = lane & 15;
  const int n0   = w << 4;

  const float* arow  = alpha + l16 * ASTRIDE + 2 * half;                      // LDS
  const float* hTrow = hT + (size_t)b * D * N + (size_t)(n0 + l16) * N + 2 * half;

  v8f c = {};
  for (int kk = 0; kk < N; kk += 16) {
    v2f a0 = *(const v2f*)(arow + kk);
    v2f a1 = *(const v2f*)(arow + kk + 4);
    v2f a2 = *(const v2f*)(arow + kk + 8);
    v2f a3 = *(const v2f*)(arow + kk + 12);
    v2f b0 = *(const v2f*)(hTrow + kk);
    v2f b1 = *(const v2f*)(hTrow + kk + 4);
    v2f b2 = *(const v2f*)(hTrow + kk + 8);
    v2f b3 = *(const v2f*)(hTrow + kk + 12);
    c = __builtin_amdgcn_wmma_f32_16x16x4_f32(false, a0, false, b0, (short)0, c, false, false);
    c = __builtin_amdgcn_wmma_f32_16x16x4_f32(false, a1, false, b1, (short)0, c, false, false);
    c = __builtin_amdgcn_wmma_f32_16x16x4_f32(false, a2, false, b2, (short)0, c, false, false);
    c = __builtin_amdgcn_wmma_f32_16x16x4_f32(false, a3, false, b3, (short)0, c, false, false);
  }
#pragma unroll
  for (int j = 0; j < 8; ++j) hp[(8 * half + j) * HPSTRIDE + n0 + l16] = c[j];
  __syncthreads();

  // ---- Phase 3: LayerNorm over D=128 per row ----
  float v[8];
  float mean = 0.f;
#pragma unroll
  for (int t = 0; t < 8; ++t) { v[t] = hp[r * HPSTRIDE + s + (t << 4)]; mean += v[t]; }
#pragma unroll
  for (int off = 8; off >= 1; off >>= 1) mean += __shfl_xor(mean, off, 16);
  mean *= (1.0f / 128.0f);
  float var = 0.f;
#pragma unroll
  for (int t = 0; t < 8; ++t) { float d = v[t] - mean; var += d * d; }
#pragma unroll
  for (int off = 8; off >= 1; off >>= 1) var += __shfl_xor(var, off, 16);
  var *= (1.0f / 128.0f);
  const float rs = rsqrtf(var + 1e-5f);

  float* orow = out + (size_t)(b * N + m0 + r) * D;
#pragma unroll
  for (int t = 0; t < 8; ++t) {
    const int d = s + (t << 4);
    orow[d] = (v[t] - mean) * rs * gamma[d] + beta[d];
  }
}

extern "C" void kernel_launch(void* const* d_in, const int* in_sizes, int n_in,
                              void* d_out, int out_size, void* d_ws, size_t ws_size,
                              hipStream_t stream) {
  (void)in_sizes; (void)n_in; (void)out_size; (void)ws_size;
  const float* x     = (const float*)d_in[0];
  const float* edge  = (const float*)d_in[1];
  const float* W     = (const float*)d_in[2];
  const float* a_src = (const float*)d_in[3];
  const float* a_dst = (const float*)d_in[4];
  const float* a_e   = (const float*)d_in[5];
  const float* gamma = (const float*)d_in[6];
  const float* beta  = (const float*)d_in[7];
  float* out = (float*)d_out;

  float* h   = (float*)d_ws;                       // 8*512*128 f32 = 2 MB
  float* hT  = h   + (size_t)BS * N * D;           // transposed copy, 2 MB
  float* s_i = hT  + (size_t)BS * N * D;           // 4096 f32
  float* s_j = s_i + (size_t)BS * N;               // 4096 f32

  // h = x @ W^T (+ transposed copy): 256 waves (8 batches x 32 row-tiles)
  k_h_gemm<<<BS * (N / 16), 32, 0, stream>>>(x, W, h, hT);
  // s_i, s_j : one wave per row
  k_scores<<<(BS * N) / 8, 256, 0, stream>>>(h, a_src, a_dst, s_i, s_j);
  // fused edge-score + softmax + alpha@h + LayerNorm
  k_attn<<<dim3(N / 16, BS), 256, 0, stream>>>(edge, hT, s_i, s_j, a_e, gamma, beta, out);
}